// CIF_Model_5970004541927
// MI455X (gfx1250) — compile-verified
//
#include <hip/hip_runtime.h>
#include <hip/hip_bf16.h>
#include <stdint.h>

#ifndef __has_builtin
#define __has_builtin(x) 0
#endif

// Problem constants (from the reference)
#define CIF_B      32
#define CIF_T      2000
#define CIF_H      512
#define CIF_MAXL   64
#define CIF_THRESH 0.95f

// Streaming-kernel tiling
#define CIF_NTC    10                       // T chunks
#define CIF_TCHUNK (CIF_T / CIF_NTC)        // 200 timesteps per chunk
#define CIF_TL     25                       // timesteps per async stage
#define CIF_NSTG   (CIF_TCHUNK / CIF_TL)    // 8 stages per chunk
#define CIF_CPB    128                      // channels per block (32 lanes x float4)
#define CIF_NHS    (CIF_H / CIF_CPB)        // 4 H-slices

#if __has_builtin(__builtin_amdgcn_global_load_async_to_lds_b128)
#define CIF_ASYNC_BUILTIN 1
#endif
#if __has_builtin(__builtin_amdgcn_s_wait_asynccnt)
#define CIF_WAITASYNC_BUILTIN 1
#endif

// Builtin parameter types: pointer-to-v4i in AS1 (global) / AS3 (LDS)
typedef int cif_v4i __attribute__((vector_size(16)));
typedef __attribute__((address_space(1))) cif_v4i* cif_gptr128;
typedef __attribute__((address_space(3))) cif_v4i* cif_lptr128;

// ---- CDNA5 async global->LDS copy (one b128 per lane) ------------------
__device__ __forceinline__ void cif_async_b128(const float* gsrc, float4* ldst)
{
#if defined(CIF_ASYNC_BUILTIN)
    __builtin_amdgcn_global_load_async_to_lds_b128(
        (cif_gptr128)(uintptr_t)(const void*)gsrc,
        (cif_lptr128)(unsigned)(uintptr_t)(void*)ldst, // low 32 bits of flat = LDS offset
        0, 0);
#else
    unsigned l = (unsigned)(uintptr_t)(void*)ldst;
    asm volatile("global_load_async_to_lds_b128 %0, %1, off"
                 :: "v"(l), "v"(gsrc) : "memory");
#endif
}

template <unsigned N>
__device__ __forceinline__ void cif_wait_async()
{
#if defined(CIF_WAITASYNC_BUILTIN)
    __builtin_amdgcn_s_wait_asynccnt(N);
#else
    asm volatile("s_wait_asynccnt %0" :: "n"(N) : "memory");
#endif
    asm volatile("" ::: "memory"); // keep ds_loads after the wait
}

// ---- Kernel 0: zero the output (we accumulate with atomics) ------------
__global__ void cif_zero_kernel(float4* __restrict__ out, int n4)
{
    int i = blockIdx.x * blockDim.x + threadIdx.x;
    if (i < n4) out[i] = make_float4(0.f, 0.f, 0.f, 0.f);
}

// ---- Kernel 1: sequential alpha scan, bit-identical to reference -------
// meta[b,t] = { w0, w1, L (int bits), fire (int bits) }
//   w0 -> weight of h_t into label L_t (= #fires before t)
//   w1 -> (fires only) remainds weight of h_t into label L_t + 1
__global__ void cif_scan_kernel(const float* __restrict__ alphas,
                                float4* __restrict__ meta,
                                int* __restrict__ nfires)
{
    int b = threadIdx.x;
    if (b >= CIF_B) return;
    const float* ap = alphas + (size_t)b * CIF_T;
    float4* mp = meta + (size_t)b * CIF_T;

    float integ = 0.0f;
    int L = 0;
    for (int t = 0; t < CIF_T; ++t) {
        float a = ap[t];
        float dist = 1.0f - integ;         // dist_completion
        integ = integ + a;
        bool fire = integ > CIF_THRESH;
        float w0 = fire ? dist : a;        // cur
        float w1 = fire ? (a - w0) : 0.0f; // remainds
        float4 m;
        m.x = w0;
        m.y = w1;
        m.z = __int_as_float(L);
        m.w = __int_as_float(fire ? 1 : 0);
        mp[t] = m;
        if (fire) { integ = integ - 1.0f; ++L; }
    }
    nfires[b] = L;
}

// ---- Kernel 2: stream hidden through LDS (async double buffer) ---------
__global__ void __launch_bounds__(32)
cif_stream_kernel(const float* __restrict__ hidden,
                  const float4* __restrict__ meta,
                  const int* __restrict__ nfires,
                  float* __restrict__ out)
{
    __shared__ float4 buf[2][CIF_TL * 32];   // 2 x 25 x 512B = 25.6 KB LDS

    const int tid = threadIdx.x;
    int bid = blockIdx.x;
    const int hs = bid % CIF_NHS;  bid /= CIF_NHS;
    const int tc = bid % CIF_NTC;  bid /= CIF_NTC;
    const int b  = bid;

    const int t0 = tc * CIF_TCHUNK;
    const int ch = hs * CIF_CPB + tid * 4;

    const float*  gbase = hidden + (size_t)b * CIF_T * CIF_H + ch;
    const float4* mrow  = meta   + (size_t)b * CIF_T;
    float*        orow  = out    + (size_t)b * CIF_MAXL * CIF_H + ch;

    // Prefetch stage 0 (one async b128 per timestep row: 32 lanes x 16B = 512B)
    {
        const float* g = gbase + (size_t)t0 * CIF_H;
        #pragma unroll
        for (int i = 0; i < CIF_TL; ++i)
            cif_async_b128(g + (size_t)i * CIF_H, &buf[0][i * 32 + tid]);
    }

    float4 acc = make_float4(0.f, 0.f, 0.f, 0.f);

    for (int s = 0; s < CIF_NSTG; ++s) {
        if (s + 1 < CIF_NSTG) {
            const float* g = gbase + (size_t)(t0 + (s + 1) * CIF_TL) * CIF_H;
            #pragma unroll
            for (int i = 0; i < CIF_TL; ++i)
                cif_async_b128(g + (size_t)i * CIF_H, &buf[(s + 1) & 1][i * 32 + tid]);
            // <=50 outstanding; async loads complete in order, so <=TL means
            // stage s's TL loads have all landed in LDS.
            cif_wait_async<CIF_TL>();
        } else {
            cif_wait_async<0>();
        }

        const float4* cb = buf[s & 1];
        const int tbase = t0 + s * CIF_TL;
        for (int i = 0; i < CIF_TL; ++i) {
            float4 m = mrow[tbase + i];           // uniform per block
            float4 h = cb[i * 32 + tid];          // ds_load_b128
            float w0 = m.x;
            if (__float_as_int(m.w)) {            // fire: label m.z completes
                int L = __float_as_int(m.z);
                if (L < CIF_MAXL) {
                    float* o = orow + (size_t)L * CIF_H;
                    atomicAdd(o + 0, fmaf(w0, h.x, acc.x));
                    atomicAdd(o + 1, fmaf(w0, h.y, acc.y));
                    atomicAdd(o + 2, fmaf(w0, h.z, acc.z));
                    atomicAdd(o + 3, fmaf(w0, h.w, acc.w));
                }
                float w1 = m.y;                   // remainds seeds next label
                acc.x = w1 * h.x; acc.y = w1 * h.y;
                acc.z = w1 * h.z; acc.w = w1 * h.w;
            } else {
                acc.x = fmaf(w0, h.x, acc.x);
                acc.y = fmaf(w0, h.y, acc.y);
                acc.z = fmaf(w0, h.z, acc.z);
                acc.w = fmaf(w0, h.w, acc.w);
            }
        }
    }

    // Chunk-boundary partial: belongs to the open label iff it fires later.
    const int t_end = t0 + CIF_TCHUNK;
    if (t_end < CIF_T) {
        int Lend = __float_as_int(mrow[t_end].z);
        if (Lend < CIF_MAXL && Lend < nfires[b]) {
            float* o = orow + (size_t)Lend * CIF_H;
            atomicAdd(o + 0, acc.x);
            atomicAdd(o + 1, acc.y);
            atomicAdd(o + 2, acc.z);
            atomicAdd(o + 3, acc.w);
        }
    }
    // t_end == T: trailing partial after the last fire is dropped (reference
    // never emits it) -> nothing to do.
}

extern "C" void kernel_launch(void* const* d_in, const int* in_sizes, int n_in,
                              void* d_out, int out_size, void* d_ws, size_t ws_size,
                              hipStream_t stream)
{
    (void)in_sizes; (void)n_in; (void)out_size; (void)ws_size;

    const float* hidden = (const float*)d_in[0];   // [B,T,H] fp32
    const float* alphas = (const float*)d_in[1];   // [B,T]   fp32
    float* out = (float*)d_out;                    // [B,64,H] fp32

    // Workspace layout: meta float4[B*T] (1,024,000 B) + nfires int[B]
    float4* meta   = (float4*)d_ws;
    int*    nfires = (int*)((char*)d_ws + sizeof(float4) * (size_t)CIF_B * CIF_T);

    // 1) zero output (accumulated via atomics)
    {
        int n4 = (CIF_B * CIF_MAXL * CIF_H) / 4;   // 262144 float4s
        int blk = 256;
        int grd = (n4 + blk - 1) / blk;
        cif_zero_kernel<<<grd, blk, 0, stream>>>((float4*)d_out, n4);
    }
    // 2) sequential alpha scan (one lane per batch)
    cif_scan_kernel<<<1, 32, 0, stream>>>(alphas, meta, nfires);
    // 3) bandwidth-bound weighted segmented reduction of hidden
    cif_stream_kernel<<<CIF_B * CIF_NTC * CIF_NHS, 32, 0, stream>>>(
        hidden, meta, nfires, out);
}